// AnteLayer_76991583748342
// MI455X (gfx1250) — compile-verified
//
#include <hip/hip_runtime.h>

typedef float v2f __attribute__((ext_vector_type(2)));
typedef float v4f __attribute__((ext_vector_type(4)));

#if defined(__has_builtin) && __has_builtin(__builtin_amdgcn_exp2f)
#define FAST_EXP2(x) __builtin_amdgcn_exp2f(x)   // v_exp_f32
#else
#define FAST_EXP2(x) __expf((x) * 0.6931471805599453f)
#endif

// gfx1250 async global->LDS path (ASYNCcnt), guarded so the file always compiles.
#if defined(__has_builtin) && \
    __has_builtin(__builtin_amdgcn_global_load_async_to_lds_b32) && \
    __has_builtin(__builtin_amdgcn_s_wait_asynccnt)
#define USE_ASYNC 1
#else
#define USE_ASYNC 0
#endif

#define AS_GLOBAL __attribute__((address_space(1)))
#define AS_LOCAL  __attribute__((address_space(3)))

#if USE_ASYNC
__device__ __forceinline__ void async_idx_load(const int* g, int* l) {
    // GLOBAL_LOAD_ASYNC_TO_LDS_B32: per-lane global->LDS DMA, tracked by ASYNCcnt.
    // Builtin signature (probe-confirmed): (global int*, local int*, imm, imm).
    __builtin_amdgcn_global_load_async_to_lds_b32(
        (AS_GLOBAL int*)g, (AS_LOCAL int*)l, /*offset=*/0, /*cpol=*/0);
}
#endif

#define BLOCK 256
#define WPB 8          // waves per block (wave32)
#define MAX_GRID 4096  // persistent grid: 4096 blocks x 8 waves = 32K waves

// Grid-stride over 256-edge chunks. Per chunk: async-staged indices (double
// buffered), b64 coordinate gathers (L2-resident table), 6x v_exp_f32,
// rank-1 outer product staged in LDS, then fully-coalesced 16B NT stores.
__global__ __launch_bounds__(BLOCK) void ante_fuzzy_kernel(
    const float* __restrict__ feat,   // [N, 8] fp32; only dims 0,1 used
    const int*   __restrict__ esrc,   // [E]
    const int*   __restrict__ edst,   // [E]
    float*       __restrict__ out,    // [E, 9]
    int E, int nChunk)
{
    __shared__ __align__(16) float stage[WPB][288];   // 9216 B output staging
#if USE_ASYNC
    __shared__ __align__(16) int sidx[2][2][BLOCK];   // [buf][src|dst][tid]
#endif

    const int tid  = threadIdx.x;
    const int lane = tid & 31;
    const int wv   = tid >> 5;
    const int Em1  = E - 1;
    const long long limitF = (long long)E * 9;        // output size in floats

#if USE_ASYNC
    // Prologue: launch DMA for this block's first chunk.
    if ((int)blockIdx.x < nChunk) {
        const int ee = min((int)blockIdx.x * BLOCK + tid, Em1);
        async_idx_load(esrc + ee, &sidx[0][0][tid]);
        async_idx_load(edst + ee, &sidx[0][1][tid]);
    }
    int buf = 0;
#endif

    for (int c = blockIdx.x; c < nChunk; c += gridDim.x) {
        int s, d;
#if USE_ASYNC
        // Issue next chunk's index DMA before consuming the current one,
        // then wait only for the 2 oldest async ops (in-order completion).
        const int cn = c + (int)gridDim.x;
        if (cn < nChunk) {
            const int en = min(cn * BLOCK + tid, Em1);
            async_idx_load(esrc + en, &sidx[buf ^ 1][0][tid]);
            async_idx_load(edst + en, &sidx[buf ^ 1][1][tid]);
            __builtin_amdgcn_s_wait_asynccnt(2);
        } else {
            __builtin_amdgcn_s_wait_asynccnt(0);
        }
        // Each thread reads only the slot its own wave's DMA wrote:
        // per-wave ASYNCcnt wait is the only sync required.
        s = sidx[buf][0][tid];
        d = sidx[buf][1][tid];
        buf ^= 1;
#else
        const int ee = min(c * BLOCK + tid, Em1);
        // Streaming single-use indices: non-temporal, keep L2 for feat table.
        s = __builtin_nontemporal_load(esrc + ee);
        d = __builtin_nontemporal_load(edst + ee);
#endif

        // Gather the 2 coordinate floats per endpoint: one b64 load each.
        // 100K x 32B = 3.2 MB table -> L2-resident, regular temporal hint.
        const v2f* f2 = (const v2f*)feat;             // feat row = 4 x v2f
        const v2f fs = f2[(size_t)s * 4];
        const v2f fd = f2[(size_t)d * 4];

        const float x1 = fd.x - fs.x;
        const float x2 = fd.y - fs.y;

        // mu = exp(-(x-c)^2/(2*0.5^2)) = exp2(-K t^2), K = 2*log2(e)
        const float K = 2.8853900817779268f;
        float m1[3], m2[3];
#pragma unroll
        for (int i = 0; i < 3; ++i) {
            const float cc = (float)(i - 1);          // centers -1, 0, +1
            const float t1 = x1 - cc;
            const float t2 = x2 - cc;
            m1[i] = FAST_EXP2(-K * t1 * t1);
            m2[i] = FAST_EXP2(-K * t2 * t2);
        }

        // Rank-1 outer product -> LDS (stride-9: gcd(9,32)=1, conflict-free).
        float* w = &stage[wv][0];
#pragma unroll
        for (int i = 0; i < 3; ++i)
#pragma unroll
            for (int j = 0; j < 3; ++j)
                w[lane * 9 + i * 3 + j] = m1[i] * m2[j];

        __syncthreads();   // order staging writes vs. readback (compiler fence)

        const long long wbaseF = ((long long)c * BLOCK + wv * 32) * 9;
        if (wbaseF + 288 <= limitF) {
            // Wave writes its contiguous 1152 B (16B-aligned) with NT b128s.
            const v4f* s4 = (const v4f*)w;
            const v4f a = s4[lane];
            const v4f b = s4[32 + lane];
            const float cl = w[256 + lane];
            v4f* o4 = (v4f*)(out + wbaseF);
            __builtin_nontemporal_store(a, o4 + lane);
            __builtin_nontemporal_store(b, o4 + 32 + lane);
            __builtin_nontemporal_store(cl, out + wbaseF + 256 + lane);
        } else {
            // Tail chunk: float-granular guarded, still coalesced.
#pragma unroll
            for (int k = 0; k < 9; ++k) {
                const long long idx = wbaseF + k * 32 + lane;
                if (idx < limitF)
                    __builtin_nontemporal_store(w[k * 32 + lane], out + idx);
            }
        }
        __syncthreads();   // stage[] reused next iteration
    }
}

extern "C" void kernel_launch(void* const* d_in, const int* in_sizes, int n_in,
                              void* d_out, int out_size, void* d_ws, size_t ws_size,
                              hipStream_t stream) {
    const float* feat = (const float*)d_in[0];   // [N*8] fp32
    const int*   esrc = (const int*)d_in[1];     // [E]
    const int*   edst = (const int*)d_in[2];     // [E]
    // d_in[3] = etypes: unused by the reference's returned tensor
    float* out = (float*)d_out;                  // [E*9] fp32
    const int E = in_sizes[1];

    const int nChunk = (E + BLOCK - 1) / BLOCK;  // E=6.4M -> 25000 chunks
    const int grid   = nChunk < MAX_GRID ? nChunk : MAX_GRID;
    ante_fuzzy_kernel<<<grid, BLOCK, 0, stream>>>(feat, esrc, edst, out, E, nChunk);
}